// Dynamics_individual_9113920602739
// MI455X (gfx1250) — compile-verified
//
#include <hip/hip_runtime.h>

// MAK dynamics: f_self[i] = 1 - x[i];  f_nbr[i,j] = -x[i] * A[i,j] * x[j]
// Pure streaming problem: ~512 MB of HBM traffic, ~0.13 GFLOP -> memory bound.
// Strategy: b128 NT loads/stores for A / f_nbr, x staged in LDS via the
// gfx1250 async global->LDS path (ASYNCcnt), ds_load_b128 for x[j] reads,
// global_prefetch on the A stream.

typedef __attribute__((ext_vector_type(4))) float v4f;
typedef __attribute__((__vector_size__(4 * sizeof(int)))) int v4i; // builtin's param type

#define THREADS 256
#define ROWS_PER_BLOCK 4
#define MAX_N4 2048 // LDS buffer holds up to 8192 floats (32 KB)

#if __has_builtin(__builtin_amdgcn_global_load_async_to_lds_b128) && \
    __has_builtin(__builtin_amdgcn_s_wait_asynccnt)
#define USE_ASYNC_LDS 1
#else
#define USE_ASYNC_LDS 0
#endif

__global__ __launch_bounds__(THREADS) void mak_dynamics_kernel(
    const float* __restrict__ x, const float* __restrict__ A,
    float* __restrict__ out, int N) {
  __shared__ v4f sx[MAX_N4]; // 32 KB: whole x vector, float4-packed

  const int tid = threadIdx.x;
  const int n4 = N >> 2;               // float4s per row
  const int iters = n4 / THREADS;      // full strips per thread (8 for N=8192)
  const int row0 = blockIdx.x * ROWS_PER_BLOCK;
  const v4f* __restrict__ x4 = (const v4f*)x;
  const bool use_lds = (n4 <= MAX_N4);

  // ---- Stage x into LDS (once per block) ----
  if (use_lds) {
#if USE_ASYNC_LDS
    for (int j = tid; j < n4; j += THREADS) {
      __builtin_amdgcn_global_load_async_to_lds_b128(
          (v4i*)(x4 + j),   // global source
          (v4i*)(sx + j),   // LDS destination
          /*offset=*/0, /*cpol=*/0);
    }
    __builtin_amdgcn_s_wait_asynccnt(0); // my transfers landed in LDS
#else
    for (int j = tid; j < n4; j += THREADS) sx[j] = x4[j];
#endif
  }
  __syncthreads(); // everyone's slice of x is now visible

  // ---- f_self for this block's rows ----
  if (tid < ROWS_PER_BLOCK) {
    int r = row0 + tid;
    if (r < N) out[r] = 1.0f - x[r];
  }

  // ---- f_nbr rows: streaming NT b128 pipeline ----
  float* __restrict__ fnbr = out + N;
  for (int rr = 0; rr < ROWS_PER_BLOCK; ++rr) {
    const int row = row0 + rr;
    if (row >= N) break;
    const float xneg = -x[row]; // uniform -> scalar load
    const v4f* __restrict__ Arow = (const v4f*)(A + (size_t)row * (size_t)N);
    v4f* __restrict__ Orow = (v4f*)(fnbr + (size_t)row * (size_t)N);

    if (use_lds) {
      // LDS path: x[j] via ds_load_b128; uniform branch, no flat selects.
#pragma unroll 4
      for (int k = 0; k < iters; ++k) {
        const int j = tid + k * THREADS;
        if (j + THREADS < n4) // prefetch next tile of the A stream
          __builtin_prefetch((const void*)(Arow + j + THREADS), 0, 0);
        v4f a = __builtin_nontemporal_load(Arow + j); // streaming, no reuse
        v4f xv = sx[j];                               // ds_load_b128
        v4f r = (xneg * a) * xv;
        __builtin_nontemporal_store(r, Orow + j);     // streaming store
      }
      for (int j = iters * THREADS + tid; j < n4; j += THREADS) { // tail
        v4f a = __builtin_nontemporal_load(Arow + j);
        v4f r = (xneg * a) * sx[j];
        __builtin_nontemporal_store(r, Orow + j);
      }
    } else {
      // Fallback: x[j] straight from global (L2-resident anyway).
      for (int j = tid; j < n4; j += THREADS) {
        v4f a = __builtin_nontemporal_load(Arow + j);
        v4f r = (xneg * a) * x4[j];
        __builtin_nontemporal_store(r, Orow + j);
      }
    }
  }
}

extern "C" void kernel_launch(void* const* d_in, const int* in_sizes, int n_in,
                              void* d_out, int out_size, void* d_ws,
                              size_t ws_size, hipStream_t stream) {
  // inputs in reference order: t (unused scalar), x [N], A [N,N]
  const float* x = (const float*)d_in[1];
  const float* A = (const float*)d_in[2];
  float* out = (float*)d_out;
  const int N = in_sizes[1];

  const int grid = (N + ROWS_PER_BLOCK - 1) / ROWS_PER_BLOCK;
  hipLaunchKernelGGL(mak_dynamics_kernel, dim3(grid), dim3(THREADS), 0, stream,
                     x, A, out, N);
}